// TAG_18631568130049
// MI455X (gfx1250) — compile-verified
//
#include <hip/hip_runtime.h>

#define NN 100000
#define NE 1600000
#define HIDD 128

typedef __attribute__((ext_vector_type(2))) float v2f;
typedef __attribute__((ext_vector_type(8))) float v8f;

static inline int cdiv(long long a, long long b) { return (int)((a + b - 1) / b); }

// ---------------- utility kernels ----------------

__global__ void k_zero4(float* __restrict__ p, int n4) {
    int i = blockIdx.x * blockDim.x + threadIdx.x;
    if (i < n4) ((float4*)p)[i] = make_float4(0.f, 0.f, 0.f, 0.f);
}

__global__ void k_deg(const int* __restrict__ dst, float* __restrict__ deg, int nE) {
    int e = blockIdx.x * blockDim.x + threadIdx.x;
    if (e < nE) atomicAdd(&deg[dst[e]], 1.0f);
}

__global__ void k_norm(float* __restrict__ d, int n) {
    int i = blockIdx.x * blockDim.x + threadIdx.x;
    if (i < n) d[i] = rsqrtf(fmaxf(d[i], 1.0f));
}

// ---------------- edge scatter (propagation hop) ----------------
// agg[dst] += h[src] * norm[src]^(SQ?2:1); D floats/node, float4 per lane.
template <int D, bool SQ>
__global__ void k_scatter(const float* __restrict__ h, const float* __restrict__ nrm,
                          const int* __restrict__ src, const int* __restrict__ dst,
                          float* __restrict__ agg, int nE) {
    constexpr int LPE = D / 4;  // lanes per edge
    long long t = (long long)blockIdx.x * blockDim.x + threadIdx.x;
    int e = (int)(t / LPE);
    int lane = (int)(t % LPE);
    if (e >= nE) return;
    int s = src[e], d = dst[e];
    float ns = nrm[s];
    if (SQ) ns *= ns;
    float4 v = ((const float4*)(h + (size_t)s * D))[lane];
    float* o = agg + (size_t)d * D + lane * 4;
    atomicAdd(o + 0, v.x * ns);
    atomicAdd(o + 1, v.y * ns);
    atomicAdd(o + 2, v.z * ns);
    atomicAdd(o + 3, v.w * ns);
}

// ---------------- fused concat-GEMM + bias + ReLU via V_WMMA_F32_16X16X4_F32 ----
// out[16 rows x 128 cols per block] = relu([f0 | norm*f1 | norm*f2] @ W + b)
// Block: 256 threads = 8 waves; wave w -> 16x16 tile at cols w*16.
template <int D>
__global__ __launch_bounds__(256) void k_gemm(const float* f0, const float* __restrict__ f1,
                                              const float* __restrict__ f2,
                                              const float* __restrict__ nrm,
                                              const float* __restrict__ W,
                                              const float* __restrict__ bias, float* out) {
    constexpr int KD = 3 * D;
    constexpr int LSTR = KD + 4;  // +4 dwords: bank-conflict-free column reads
    __shared__ float lds[16 * LSTR];
    const int row0 = blockIdx.x * 16;
    const int tid = threadIdx.x;

    // Stage A tile [16 x 3D] into LDS, fusing per-row norm scaling of segs 1,2.
    const float* segs[3] = {f0, f1, f2};
#pragma unroll
    for (int s = 0; s < 3; ++s) {
        const float* sp = segs[s];
        for (int idx = tid; idx < 16 * (D / 4); idx += 256) {
            int r = idx / (D / 4);
            int c = (idx % (D / 4)) * 4;
            int row = row0 + r;
            float4 v = *(const float4*)(sp + (size_t)row * D + c);
            float sc = (s == 0) ? 1.0f : nrm[row];
            float* l = &lds[r * LSTR + s * D + c];
            l[0] = v.x * sc; l[1] = v.y * sc; l[2] = v.z * sc; l[3] = v.w * sc;
        }
    }
    __syncthreads();

    const int wave = tid >> 5;
    const int lane = tid & 31;
    const int ncol = wave * 16;
    const int n = lane & 15;
    const int half = lane >> 4;   // lanes 16-31 carry K+2,K+3 (A) / rows K+2,K+3 (B)
    const int koff = half * 2;
    const float* wp = W + ncol + n;

    v8f acc = {};
#pragma unroll 4
    for (int k = 0; k < KD; k += 4) {
        v2f a, b;
        a.x = lds[n * LSTR + k + koff];
        a.y = lds[n * LSTR + k + koff + 1];
        b.x = wp[(size_t)(k + koff) * HIDD];
        b.y = wp[(size_t)(k + koff + 1) * HIDD];
        acc = __builtin_amdgcn_wmma_f32_16x16x4_f32(false, a, false, b, (short)0, acc,
                                                    false, false);
    }

    float bb = bias[ncol + n];
#pragma unroll
    for (int j = 0; j < 8; ++j) {
        int m = j + half * 8;  // C/D layout: VGPR j -> M = j (+8 for upper half-lanes)
        float v = acc[j] + bb;
        out[(size_t)(row0 + m) * HIDD + ncol + n] = fmaxf(v, 0.0f);
    }
}

// ---------------- orchestration ----------------
extern "C" void kernel_launch(void* const* d_in, const int* in_sizes, int n_in, void* d_out,
                              int out_size, void* d_ws, size_t ws_size, hipStream_t stream) {
    const float* x  = (const float*)d_in[0];
    const int* src  = (const int*)d_in[1];
    const int* dst  = (const int*)d_in[2];
    const float* W1 = (const float*)d_in[3];
    const float* b1 = (const float*)d_in[4];
    const float* W2 = (const float*)d_in[5];
    const float* b2 = (const float*)d_in[6];
    const float* W3 = (const float*)d_in[7];
    const float* b3 = (const float*)d_in[8];
    float* outp = (float*)d_out;

    float* ws  = (float*)d_ws;
    float* nrm = ws;                             // NN
    float* f1  = nrm + NN;                       // NN*128
    float* f2  = f1 + (size_t)NN * 128;          // NN*128
    float* h1  = f2 + (size_t)NN * 128;          // NN*128  (layer1 output)

    const int T = 256;
    const int gRow = NN / 16;  // 6250, exact

    // degree -> norm = rsqrt(max(deg,1))
    k_zero4<<<cdiv(NN / 4, T), T, 0, stream>>>(nrm, NN / 4);
    k_deg<<<cdiv(NE, T), T, 0, stream>>>(dst, nrm, NE);
    k_norm<<<cdiv(NN, T), T, 0, stream>>>(nrm, NN);

    // ---- layer 1 (d=32) ----
    k_zero4<<<cdiv((size_t)NN * 32 / 4, T), T, 0, stream>>>(f1, NN * 32 / 4);
    k_scatter<32, false><<<cdiv((long long)NE * 8, T), T, 0, stream>>>(x, nrm, src, dst, f1, NE);
    k_zero4<<<cdiv((size_t)NN * 32 / 4, T), T, 0, stream>>>(f2, NN * 32 / 4);
    k_scatter<32, true><<<cdiv((long long)NE * 8, T), T, 0, stream>>>(f1, nrm, src, dst, f2, NE);
    k_gemm<32><<<gRow, T, 0, stream>>>(x, f1, f2, nrm, W1, b1, h1);

    // ---- layer 2 (d=128), output to d_out (used as h2) ----
    k_zero4<<<cdiv((size_t)NN * 128 / 4, T), T, 0, stream>>>(f1, NN * 128 / 4);
    k_scatter<128, false><<<cdiv((long long)NE * 32, T), T, 0, stream>>>(h1, nrm, src, dst, f1, NE);
    k_zero4<<<cdiv((size_t)NN * 128 / 4, T), T, 0, stream>>>(f2, NN * 128 / 4);
    k_scatter<128, true><<<cdiv((long long)NE * 32, T), T, 0, stream>>>(f1, nrm, src, dst, f2, NE);
    k_gemm<128><<<gRow, T, 0, stream>>>(h1, f1, f2, nrm, W2, b2, outp);

    // ---- layer 3 (d=128), reads d_out as f0, writes d_out in place (per-block
    // row-tile exclusivity: all reads of a block's 16 rows complete before its stores) ----
    k_zero4<<<cdiv((size_t)NN * 128 / 4, T), T, 0, stream>>>(f1, NN * 128 / 4);
    k_scatter<128, false><<<cdiv((long long)NE * 32, T), T, 0, stream>>>(outp, nrm, src, dst, f1, NE);
    k_zero4<<<cdiv((size_t)NN * 128 / 4, T), T, 0, stream>>>(f2, NN * 128 / 4);
    k_scatter<128, true><<<cdiv((long long)NE * 32, T), T, 0, stream>>>(f1, nrm, src, dst, f2, NE);
    k_gemm<128><<<gRow, T, 0, stream>>>(outp, f1, f2, nrm, W3, b3, outp);
}